// KalmanFilter_13494787244416
// MI455X (gfx1250) — compile-verified
//
#include <hip/hip_runtime.h>
#include <stdint.h>

typedef __attribute__((ext_vector_type(2))) float v2f;
typedef __attribute__((ext_vector_type(8))) float v8f;

#define TLEN  131072
#define NCONV 574            // exact Riccati + exact est steps (t = 2 .. 575)
#define HEAD  576            // 2 + NCONV; (TLEN-HEAD) % 64 == 0
#define CHUNK 64
#define NCH   2039           // (TLEN - HEAD) / CHUNK
#define AMC   2048           // padded chunk-column count (multiple of 16)
#define KDIM  780            // 12*(CHUNK+1): [e0 | a_0 .. a_63] stacked
#define WROWS 768            // 12*CHUNK output rows per chunk
#define GEMM_WAVES 8

// ---------------------------------------------------------------------------
// Phase A: exact serial head. One workgroup runs the reference recursion
// (block-structured predict, 6x6 Gauss-Jordan inverse, gain, covariance
// update, estimate update) for t = 2 .. HEAD-1, writing full 12-dim estimates
// into est_all. Then it emits K_ss, M = (I-KH)F, and the powers G_j = M^j
// (j = 0..64) needed by the bulk affine-scan phase.
// ---------------------------------------------------------------------------
__global__ void kf_phaseA(const float* __restrict__ meas,
                          const float* __restrict__ Q,
                          const float* __restrict__ Rm,
                          const float* __restrict__ P0,
                          float* __restrict__ est_all,
                          float* __restrict__ Gpow,
                          float* __restrict__ Kss)
{
    __shared__ float P[12][12], Pp[12][12], Kf[12][6];
    __shared__ float Aug[6][12], fac[6];
    __shared__ float est[12], estn[12], innov[6];
    __shared__ float Msh[12][12], Ga[12][12], Gb[12][12];

    const int tid = threadIdx.x;
    const int i = tid / 12, j = tid % 12;   // valid when tid < 144

    if (tid < 144) P[i][j] = P0[tid];
    if (tid < 12) {
        // row0 = [pm0 | 0], row1 = [pm1 | pm0]; scan carry starts at row1
        est_all[0 * 12 + tid] = (tid < 6) ? meas[tid] : 0.0f;
        float e1 = (tid < 6) ? meas[6 + tid] : meas[tid - 6];
        est_all[1 * 12 + tid] = e1;
        est[tid] = e1;
    }
    __syncthreads();

    for (int t = 2; t < HEAD; ++t) {
        // ---- predict: Pp = F P F^T + Qaug (pure block adds) ----
        if (tid < 144) {
            float v;
            if (i < 6 && j < 6)
                v = 4.0f * P[i][j] - 2.0f * P[i + 6][j] - 2.0f * P[i][j + 6]
                    + P[i + 6][j + 6] + Q[i * 6 + j];
            else if (i < 6)
                v = 2.0f * P[i][j - 6] - P[i + 6][j - 6];
            else if (j < 6)
                v = 2.0f * P[i - 6][j] - P[i - 6][j + 6];
            else
                v = P[i - 6][j - 6];
            Pp[i][j] = v;
        }
        __syncthreads();

        // ---- S = Pp[0:6,0:6] + R ; Gauss-Jordan invert [S | I] ----
        if (tid < 72) {
            int r = tid / 12, c = tid % 12;
            Aug[r][c] = (c < 6) ? (Pp[r][c] + Rm[r * 6 + c])
                                : ((c - 6 == r) ? 1.0f : 0.0f);
        }
        __syncthreads();
        for (int kp = 0; kp < 6; ++kp) {
            float pv = 0.0f;
            if (tid < 12) pv = Aug[kp][kp];
            __syncthreads();
            if (tid < 12) Aug[kp][tid] = Aug[kp][tid] * (1.0f / pv);
            if (tid < 6 && tid != kp) fac[tid] = Aug[tid][kp];
            __syncthreads();
            if (tid < 72) {
                int r = tid / 12, c = tid % 12;
                if (r != kp) Aug[r][c] -= fac[r] * Aug[kp][c];
            }
            __syncthreads();
        }

        // ---- K = Pp[:, :6] * Sinv ----
        if (tid < 72) {
            int r = tid / 6, c = tid % 6;
            float s = 0.0f;
            for (int q = 0; q < 6; ++q) s += Pp[r][q] * Aug[q][6 + c];
            Kf[r][c] = s;
        }
        __syncthreads();

        // ---- estimate update ----
        if (tid < 6)
            innov[tid] = meas[t * 6 + tid] - (2.0f * est[tid] - est[tid + 6]);
        __syncthreads();
        if (tid < 12) {
            float pr = (tid < 6) ? (2.0f * est[tid] - est[tid + 6]) : est[tid - 6];
            float s = pr;
            for (int q = 0; q < 6; ++q) s += Kf[tid][q] * innov[q];
            estn[tid] = s;
            est_all[(size_t)t * 12 + tid] = s;
        }
        // ---- P = Pp - K * Pp[0:6, :] ----
        if (tid < 144) {
            float s = Pp[i][j];
            for (int q = 0; q < 6; ++q) s -= Kf[i][q] * Pp[q][j];
            P[i][j] = s;
        }
        __syncthreads();
        if (tid < 12) est[tid] = estn[tid];
        __syncthreads();
    }

    // ---- steady-state gain, M = F - [K 0]F, matrix powers G_j = M^j ----
    if (tid < 72) Kss[tid] = Kf[tid / 6][tid % 6];
    if (tid < 144) {
        float Fij;
        if (i < 6) Fij = ((j == i) ? 2.0f : 0.0f) + ((j == i + 6) ? -1.0f : 0.0f);
        else       Fij = (j == i - 6) ? 1.0f : 0.0f;
        float corr = (j < 6) ? 2.0f * Kf[i][j] : -Kf[i][j - 6];
        float m = Fij - corr;
        Msh[i][j] = m;
        Ga[i][j] = m;
        Gpow[0 * 144 + tid] = (i == j) ? 1.0f : 0.0f;
        Gpow[1 * 144 + tid] = m;
    }
    __syncthreads();
    for (int p = 2; p <= CHUNK; ++p) {
        if (tid < 144) {
            float s = 0.0f;
            for (int q = 0; q < 12; ++q) s += Ga[i][q] * Msh[q][j];
            Gb[i][j] = s;
            Gpow[(size_t)p * 144 + tid] = s;
        }
        __syncthreads();
        if (tid < 144) Ga[i][j] = Gb[i][j];
        __syncthreads();
    }
}

// ---------------------------------------------------------------------------
// Build Wbig (768 x 780), block-Toeplitz in the G_j powers:
//   row r = 12*j + p, col 0..11        -> G_{j+1}[p][q]
//   col 12*(i+1)+q                     -> (i<=j) ? G_{j-i}[p][q] : 0
// ---------------------------------------------------------------------------
__global__ void kf_wfill(const float* __restrict__ Gpow, float* __restrict__ Wb)
{
    int idx = blockIdx.x * blockDim.x + threadIdx.x;
    if (idx >= WROWS * KDIM) return;
    int r = idx / KDIM, c = idx % KDIM;
    int jj = r / 12, p = r % 12;
    float v;
    if (c < 12) {
        v = Gpow[(size_t)(jj + 1) * 144 + p * 12 + c];
    } else {
        int ii = (c - 12) / 12, q = (c - 12) % 12;
        v = (ii <= jj) ? Gpow[(size_t)(jj - ii) * 144 + p * 12 + q] : 0.0f;
    }
    Wb[idx] = v;
}

// ---------------------------------------------------------------------------
// Build U (780 x 2048): rows 0..11 zeroed (chunk incoming states, filled by
// the inter-chunk scan); rows 12*(i+1)+q hold a_i[q] = (K_ss * meas_t)[q]
// with t = HEAD + 64*c + i. Padded columns are zero.
// ---------------------------------------------------------------------------
__global__ void kf_amatfill(const float* __restrict__ meas,
                            const float* __restrict__ Kss,
                            float* __restrict__ Amat)
{
    int idx = blockIdx.x * blockDim.x + threadIdx.x;
    if (idx >= KDIM * AMC) return;
    int r = idx / AMC, c = idx % AMC;
    float v = 0.0f;
    if (r >= 12 && c < NCH) {
        int ii = (r - 12) / 12, q = (r - 12) % 12;
        int t = HEAD + CHUNK * c + ii;
        float s = 0.0f;
        for (int p = 0; p < 6; ++p) s += Kss[q * 6 + p] * meas[(size_t)t * 6 + p];
        v = s;
    }
    Amat[idx] = v;
}

// ---------------------------------------------------------------------------
// WMMA f32 GEMM: C[16x16 tiles] = Wb(768x780) x Amat(780x2048).
// All 8 waves of a block share one 16-row A panel (nColTiles % 8 == 0), so
// the panel (16x780 f32 = 48.75 KB << 320 KB LDS) is staged cooperatively
// with gfx1250 async-to-LDS loads (ASYNCcnt), then each wave runs the K loop
// reading A fragments from LDS (ds_load_b64) and B columns from global,
// accumulating with V_WMMA_F32_16X16X4_F32.
// A-frag (16x4 f32): lanes 0-15 rows, VGPR0/1 = K{0,1} (lanes 0-15) /
// K{2,3} (lanes 16-31). B-frag mirrored on K halves. D layout: VGPR r ->
// row r (lanes 0-15) / row r+8 (lanes 16-31).
// ---------------------------------------------------------------------------
__global__ __launch_bounds__(256) void kf_gemm(const float* __restrict__ Wb,
                        const float* __restrict__ Bm,
                        float* __restrict__ Cm,
                        int rowTileStart, int nColTiles, int nTiles, int crowOff)
{
    __shared__ float As[16 * KDIM];   // A panel for this block's row tile

    // row tile is uniform across the block
    int rt = rowTileStart + (blockIdx.x * GEMM_WAVES) / nColTiles;

    // ---- cooperative async stage of the A panel: 3120 x b128 transfers ----
    {
        unsigned ldsBase = (unsigned)(uintptr_t)(&As[0]);   // addr[31:0] = LDS offset
        for (int idx = threadIdx.x; idx < 16 * (KDIM / 4); idx += 256) {
            int row = idx / (KDIM / 4);
            int kc  = idx % (KDIM / 4);
            unsigned goff  = (unsigned)(((rt * 16 + row) * KDIM + kc * 4) * 4);
            unsigned laddr = ldsBase + (unsigned)((row * KDIM + kc * 4) * 4);
            asm volatile("global_load_async_to_lds_b128 %0, %1, %2"
                         :: "v"(laddr), "v"(goff), "s"(Wb) : "memory");
        }
        asm volatile("s_wait_asynccnt 0x0" ::: "memory");
    }
    __syncthreads();

    int wave = blockIdx.x * GEMM_WAVES + (threadIdx.x >> 5);
    if (wave < nTiles) {
        int ct   = wave % nColTiles;
        int lane = threadIdx.x & 31;
        int mrow = lane & 15;
        int half = lane >> 4;
        int col  = ct * 16 + mrow;
        const float* Arow = &As[mrow * KDIM];

        v8f acc = {};
        for (int k = 0; k < KDIM; k += 4) {
            v2f a, b;
            a.x = Arow[k + 2 * half];
            a.y = Arow[k + 2 * half + 1];
            b.x = Bm[(size_t)(k + 2 * half) * AMC + col];
            b.y = Bm[(size_t)(k + 2 * half + 1) * AMC + col];
            acc = __builtin_amdgcn_wmma_f32_16x16x4_f32(
                false, a, false, b, (short)0, acc, false, false);
        }
        for (int r = 0; r < 8; ++r) {
            int orow = rt * 16 + r + 8 * half - crowOff;
            Cm[(size_t)orow * AMC + col] = acc[r];
        }
    }
}

// ---------------------------------------------------------------------------
// Inter-chunk serial scan (2039 12x12 matvecs), single wave32 so every
// __syncthreads is a hardware NOP (single-wave workgroup) and same-wave LDS
// ordering carries the dependence:
//   e_{c+1} = G64 * e_c + r_c,  r_c = Rbuf rows 4..15 col c
// and deposits each chunk's incoming state e_c into Amat rows 0..11.
// ---------------------------------------------------------------------------
__global__ __launch_bounds__(32) void kf_scan(const float* __restrict__ Gpow,
                        const float* __restrict__ Rbuf,
                        const float* __restrict__ est_all,
                        float* __restrict__ Amat)
{
    __shared__ float G64[12][12];
    __shared__ float e[12], en[12];
    int tid = threadIdx.x;
    for (int x = tid; x < 144; x += 32)
        G64[x / 12][x % 12] = Gpow[(size_t)CHUNK * 144 + x];
    if (tid < 12) e[tid] = est_all[(size_t)(HEAD - 1) * 12 + tid];
    __syncthreads();
    for (int c = 0; c < NCH; ++c) {
        if (tid < 12) {
            Amat[(size_t)tid * AMC + c] = e[tid];
            float s = Rbuf[(size_t)(4 + tid) * AMC + c];
            for (int q = 0; q < 12; ++q) s += G64[tid][q] * e[q];
            en[tid] = s;
        }
        __syncthreads();
        if (tid < 12) e[tid] = en[tid];
        __syncthreads();
    }
}

// Scatter GEMM2 output back into the time-major est array.
__global__ void kf_b3(const float* __restrict__ OutA, float* __restrict__ est_all)
{
    int idx = blockIdx.x * blockDim.x + threadIdx.x;
    if (idx >= (TLEN - HEAD) * 12) return;
    int s = idx / 12, q = idx % 12;
    int c = s / CHUNK, jj = s % CHUNK;
    est_all[(size_t)(HEAD + s) * 12 + q] = OutA[(size_t)(12 * jj + q) * AMC + c];
}

// Elementwise output assembly: pose_estimate, pose_prediction, velocity.
__global__ void kf_out(const float* __restrict__ est_all, float* __restrict__ out)
{
    int idx = blockIdx.x * blockDim.x + threadIdx.x;
    if (idx >= TLEN * 6) return;
    int t = idx / 6, k = idx % 6;
    float e0 = est_all[(size_t)t * 12 + k];
    out[idx] = e0;                                        // pose_estimate
    float pr;
    if (t < 2) pr = e0;
    else pr = 2.0f * est_all[(size_t)(t - 1) * 12 + k]
              - est_all[(size_t)(t - 1) * 12 + k + 6];
    out[(size_t)TLEN * 6 + idx] = pr;                     // pose_prediction
    if (t >= 1 && t <= TLEN - 2) {                        // velocity rows 1..T-2
        float v = e0 - est_all[(size_t)t * 12 + k + 6];
        out[(size_t)TLEN * 12 + (size_t)(t - 1) * 6 + k] = v;
    }
}

extern "C" void kernel_launch(void* const* d_in, const int* in_sizes, int n_in,
                              void* d_out, int out_size, void* d_ws, size_t ws_size,
                              hipStream_t stream)
{
    const float* meas = (const float*)d_in[0];   // (T,6)
    const float* Q    = (const float*)d_in[1];   // (6,6)
    const float* Rm   = (const float*)d_in[2];   // (6,6)
    const float* P0   = (const float*)d_in[3];   // (12,12)
    float* out = (float*)d_out;

    float* w       = (float*)d_ws;
    float* est_all = w;                                   // T*12
    float* Gpow    = est_all + (size_t)TLEN * 12;         // 65*144
    float* Kss     = Gpow + 65 * 144;                     // 72
    float* Wb      = Kss + 72;                            // 768*780
    float* Amat    = Wb + (size_t)WROWS * KDIM;           // 780*2048
    float* Rbuf    = Amat + (size_t)KDIM * AMC;           // 16*2048
    float* OutA    = Rbuf + (size_t)16 * AMC;             // 768*2048

    // Phase A: exact serial head + steady-state M, K, powers
    kf_phaseA<<<1, 256, 0, stream>>>(meas, Q, Rm, P0, est_all, Gpow, Kss);

    // Build Wbig and the chunk input matrix
    kf_wfill<<<(WROWS * KDIM + 255) / 256, 256, 0, stream>>>(Gpow, Wb);
    kf_amatfill<<<(KDIM * AMC + 255) / 256, 256, 0, stream>>>(meas, Kss, Amat);

    // GEMM1 (WMMA): last block-row only -> chunk-local offsets r_c
    {
        int nTiles = 1 * (AMC / 16);
        kf_gemm<<<(nTiles + 7) / 8, 256, 0, stream>>>(Wb, Amat, Rbuf,
                                                      WROWS / 16 - 1, AMC / 16,
                                                      nTiles, WROWS - 16);
    }
    // Inter-chunk serial scan fills chunk incoming states into Amat rows 0..11
    kf_scan<<<1, 32, 0, stream>>>(Gpow, Rbuf, est_all, Amat);

    // GEMM2 (WMMA): all within-chunk prefixes at once
    {
        int nTiles = (WROWS / 16) * (AMC / 16);
        kf_gemm<<<(nTiles + 7) / 8, 256, 0, stream>>>(Wb, Amat, OutA,
                                                      0, AMC / 16, nTiles, 0);
    }
    // Scatter to time-major estimates, then assemble the three outputs
    kf_b3<<<((TLEN - HEAD) * 12 + 255) / 256, 256, 0, stream>>>(OutA, est_all);
    kf_out<<<(TLEN * 6 + 255) / 256, 256, 0, stream>>>(est_all, out);
}